// LieAlgebraPositionedRectangles_56925496541962
// MI455X (gfx1250) — compile-verified
//
#include <hip/hip_runtime.h>
#include <stdint.h>

// One lane = one 4x4 matrix exponential, computed in f64 (matches reference's
// float64 expm), exploiting the zero bottom row of A (implicit row3 = [0,0,0,1]).
// Input staged to LDS via gfx1250 async data mover (global_load_async_to_lds_b128).

#define BLOCK 256
#define FLOATS_PER_BLOCK (BLOCK * 7)          // 1792 floats = 7168 B
#define CHUNKS_PER_BLOCK (FLOATS_PER_BLOCK/4) // 448 b128 chunks

__global__ __launch_bounds__(BLOCK) void lie_expm_kernel(const float* __restrict__ in,
                                                         float* __restrict__ out,
                                                         int n) {
    __shared__ float s_in[FLOATS_PER_BLOCK];

    const int tid = threadIdx.x;
    const long long blockElemBase = (long long)blockIdx.x * BLOCK;
    const long long floatBase     = blockElemBase * 7;
    const long long totalFloats   = (long long)n * 7;   // 7,000,000 (divisible by 4)
    const unsigned long long base = (unsigned long long)(uintptr_t)in;

    // ---- Stage this block's inputs into LDS with async 128-bit copies ----
    // 448 chunks of 16 B; 256 lanes -> 2 rounds. Chunk starts are 4-float
    // aligned and totalFloats % 4 == 0, so every chunk is fully in- or out-of-range.
    #pragma unroll
    for (int it = 0; it < 2; ++it) {
        int c = it * BLOCK + tid;
        if (c < CHUNKS_PER_BLOCK) {
            long long g = floatBase + (long long)c * 4;   // global float index
            if (g < totalFloats) {
                uint32_t ldsaddr = (uint32_t)(uintptr_t)(s_in + c * 4); // low 32 bits = LDS offset
                uint32_t voff    = (uint32_t)(g * 4);                   // byte offset from base
                asm volatile("global_load_async_to_lds_b128 %0, %1, %2"
                             :
                             : "v"(ldsaddr), "v"(voff), "s"(base)
                             : "memory");
            }
        }
    }
    asm volatile("s_wait_asynccnt 0x0" ::: "memory");
    __syncthreads();

    const long long elem = blockElemBase + tid;
    if (elem >= n) return;

    // ---- Load 7 coefficients (stride 7 in LDS: gcd(7,64)=1 -> conflict-free) ----
    const float* vv = s_in + tid * 7;
    const double v0 = (double)vv[0], v1 = (double)vv[1], v2 = (double)vv[2];
    const double v3 = (double)vv[3], v4 = (double)vv[4], v5 = (double)vv[5];
    const double v6 = (double)vv[6];

    // A (rows 0..2; row 3 is zero):
    //   [  v6  -v5   v4  v0 ]
    //   [  v5  -v6  -v3  v1 ]
    //   [ -v4   v3    0  v2 ]
    double a[12];
    a[0] =  v6; a[1]  = -v5; a[2]  =  v4; a[3]  = v0;
    a[4] =  v5; a[5]  = -v6; a[6]  = -v3; a[7]  = v1;
    a[8] = -v4; a[9]  =  v3; a[10] = 0.0; a[11] = v2;

    // inf-norm over the 3 nonzero rows
    double r0 = fabs(a[0]) + fabs(a[1]) + fabs(a[2])  + fabs(a[3]);
    double r1 = fabs(a[4]) + fabs(a[5]) + fabs(a[6])  + fabs(a[7]);
    double r2 = fabs(a[8]) + fabs(a[9]) + fabs(a[10]) + fabs(a[11]);
    double nrm = fmax(r0, fmax(r1, r2));

    // scaling: s so that nrm * 2^-s in [2^-6, 2^-5]
    long long nb = __double_as_longlong(nrm);
    int e = (int)((nb >> 52) & 0x7FF) - 1022;     // nrm in [2^(e-1), 2^e)
    int s = e + 5;
    if (s < 0)  s = 0;
    if (s > 60) s = 60;
    double sc = __longlong_as_double((long long)(1023 - s) << 52); // 2^-s
    #pragma unroll
    for (int i = 0; i < 12; ++i) a[i] *= sc;

    // ---- Taylor-Horner, order 9: X = I + A(I + A/2(I + ... (I + A/9))) ----
    // X keeps implicit bottom row [0,0,0,1]; A has zero bottom row.
    double x[12];
    {
        const double c9 = 1.0 / 9.0;
        #pragma unroll
        for (int i = 0; i < 3; ++i)
            #pragma unroll
            for (int j = 0; j < 4; ++j)
                x[i*4+j] = ((i == j) ? 1.0 : 0.0) + a[i*4+j] * c9;
    }
    const double rc[8] = {1.0, 1.0/2.0, 1.0/3.0, 1.0/4.0,
                          1.0/5.0, 1.0/6.0, 1.0/7.0, 1.0/8.0};
    #pragma unroll
    for (int k = 8; k >= 1; --k) {
        double p[12];
        #pragma unroll
        for (int i = 0; i < 3; ++i) {
            #pragma unroll
            for (int j = 0; j < 4; ++j) {
                p[i*4+j] = a[i*4+0] * x[0*4+j]
                         + a[i*4+1] * x[1*4+j]
                         + a[i*4+2] * x[2*4+j];
            }
            p[i*4+3] += a[i*4+3];          // A[:,3] * X[3][3](=1)
        }
        const double c = rc[k-1];
        #pragma unroll
        for (int i = 0; i < 3; ++i)
            #pragma unroll
            for (int j = 0; j < 4; ++j)
                x[i*4+j] = ((i == j) ? 1.0 : 0.0) + c * p[i*4+j];
    }

    // ---- Repeated squaring (implicit row3 = [0,0,0,1]) ----
    for (int t = 0; t < s; ++t) {
        double r[12];
        #pragma unroll
        for (int i = 0; i < 3; ++i) {
            #pragma unroll
            for (int j = 0; j < 3; ++j) {
                r[i*4+j] = x[i*4+0] * x[0*4+j]
                         + x[i*4+1] * x[1*4+j]
                         + x[i*4+2] * x[2*4+j];
            }
            r[i*4+3] = x[i*4+0] * x[0*4+3]
                     + x[i*4+1] * x[1*4+3]
                     + x[i*4+2] * x[2*4+3]
                     + x[i*4+3];
        }
        #pragma unroll
        for (int i = 0; i < 12; ++i) x[i] = r[i];
    }

    // T[3][3] == 1 exactly (zero bottom row of A), so normalization is identity.
    float4* o = reinterpret_cast<float4*>(out + (size_t)elem * 16);
    o[0] = make_float4((float)x[0], (float)x[1],  (float)x[2],  (float)x[3]);
    o[1] = make_float4((float)x[4], (float)x[5],  (float)x[6],  (float)x[7]);
    o[2] = make_float4((float)x[8], (float)x[9],  (float)x[10], (float)x[11]);
    o[3] = make_float4(0.0f, 0.0f, 0.0f, 1.0f);
}

extern "C" void kernel_launch(void* const* d_in, const int* in_sizes, int n_in,
                              void* d_out, int out_size, void* d_ws, size_t ws_size,
                              hipStream_t stream) {
    (void)n_in; (void)out_size; (void)d_ws; (void)ws_size;
    const float* in = (const float*)d_in[0];
    float* out = (float*)d_out;
    const int n = in_sizes[0] / 7;                 // B = 1,000,000
    const int grid = (n + BLOCK - 1) / BLOCK;
    lie_expm_kernel<<<grid, BLOCK, 0, stream>>>(in, out, n);
}